// ScaledDotProductAttention_40767829574593
// MI455X (gfx1250) — compile-verified
//
#include <hip/hip_runtime.h>
#include <math.h>

typedef __attribute__((ext_vector_type(2))) float v2f;
typedef __attribute__((ext_vector_type(8))) float v8f;

#define B_    64
#define S_    729
#define D_    64
#define NQT   46      // ceil(729/16)
#define WAVES 4

__global__ __launch_bounds__(WAVES * 32)
void attn_tanh_wmma_kernel(const float* __restrict__ Q,
                           const float* __restrict__ Kp,
                           const float* __restrict__ V,
                           const int*   __restrict__ Mk,
                           float* __restrict__ Out,    // [B,S,D]
                           float* __restrict__ Attn)   // [B,S,S]
{
    __shared__ float lds[WAVES][16][17];   // padded to dodge bank conflicts

    const int lane = threadIdx.x & 31;
    const int wave = threadIdx.x >> 5;
    const int qtile = blockIdx.x * WAVES + wave;
    if (qtile >= NQT) return;              // whole-wave uniform exit
    const int b = blockIdx.y;

    const int r = lane & 15;               // row-in-tile / col-in-tile index
    const int h = lane >> 4;               // lane half (selects K pair in frags)
    const int qbase = qtile * 16;

    // ---- A fragments for Q tile: aQ[t] covers inner-dim chunk d0 = 4*t ----
    const int qrow = min(qbase + r, S_ - 1);
    const float* qptr = Q + ((size_t)b * S_ + qrow) * D_ + 2 * h;
    v2f aQ[16];
#pragma unroll
    for (int t = 0; t < 16; ++t)
        aQ[t] = *(const v2f*)(qptr + t * 4);

    // ---- output accumulators: 4 x (16x16 f32 C tiles) = 16x64 ----
    v8f acc[4];
#pragma unroll
    for (int nd = 0; nd < 4; ++nd)
        acc[nd] = (v8f){0.f, 0.f, 0.f, 0.f, 0.f, 0.f, 0.f, 0.f};

    for (int kt = 0; kt < NQT; ++kt) {
        const int kbase = kt * 16;

        // ---- scores tile: 16 WMMAs over D=64 ----
        const int krow = min(kbase + r, S_ - 1);
        const float* kptr = Kp + ((size_t)b * S_ + krow) * D_ + 2 * h;
        v8f c = (v8f){0.f, 0.f, 0.f, 0.f, 0.f, 0.f, 0.f, 0.f};
#pragma unroll
        for (int t = 0; t < 16; ++t) {
            v2f bK = *(const v2f*)(kptr + t * 4);
            c = __builtin_amdgcn_wmma_f32_16x16x4_f32(
                    false, aQ[t], false, bK, (short)0, c, false, false);
        }

        // ---- elementwise: scale, mask, tanh, diag-zero; store attention ----
        // mask / attention are zero-reuse 136MB streams: use non-temporal
        // hints so they don't evict the reused Q/K/V working set from L2.
        const int k = kbase + r;
#pragma unroll
        for (int j = 0; j < 8; ++j) {
            const int q = qbase + j + 8 * h;
            const float s = c[j] * 0.125f;                 // 1/sqrt(64)
            const int   m = __builtin_nontemporal_load(
                Mk + ((size_t)b * S_ + min(q, S_ - 1)) * S_ + min(k, S_ - 1));
            float a = (m == 0) ? -1.0f : tanhf(s);         // tanh(-1e9) == -1
            if (k >= S_ || k == q) a = 0.0f;               // fringe + diagonal
            c[j] = a;
            if (q < S_ && k < S_)
                __builtin_nontemporal_store(
                    a, Attn + (size_t)b * S_ * S_ + (size_t)q * S_ + k);
            lds[wave][j + 8 * h][r] = a;                   // C-layout -> LDS
        }

        // ---- re-load attention tile in A-layout (4 chunks of K=4) ----
        v2f aA[4];
#pragma unroll
        for (int kc = 0; kc < 4; ++kc) {
            aA[kc][0] = lds[wave][r][kc * 4 + 2 * h];
            aA[kc][1] = lds[wave][r][kc * 4 + 2 * h + 1];
        }

        // ---- out += attn_tile @ V_tile : 4 N-tiles x 4 K-chunks ----
#pragma unroll
        for (int nd = 0; nd < 4; ++nd) {
#pragma unroll
            for (int kc = 0; kc < 4; ++kc) {
                const int kr0 = min(kbase + kc * 4 + 2 * h,     S_ - 1);
                const int kr1 = min(kbase + kc * 4 + 2 * h + 1, S_ - 1);
                v2f bV;
                bV[0] = V[((size_t)b * S_ + kr0) * D_ + nd * 16 + r];
                bV[1] = V[((size_t)b * S_ + kr1) * D_ + nd * 16 + r];
                acc[nd] = __builtin_amdgcn_wmma_f32_16x16x4_f32(
                              false, aA[kc], false, bV, (short)0, acc[nd],
                              false, false);
            }
        }
    }

    // ---- store out tile (coalesced, fringe-guarded) ----
#pragma unroll
    for (int nd = 0; nd < 4; ++nd) {
#pragma unroll
        for (int j = 0; j < 8; ++j) {
            const int q = qbase + j + 8 * h;
            if (q < S_)
                Out[((size_t)b * S_ + q) * D_ + nd * 16 + r] = acc[nd][j];
        }
    }
}

extern "C" void kernel_launch(void* const* d_in, const int* in_sizes, int n_in,
                              void* d_out, int out_size, void* d_ws, size_t ws_size,
                              hipStream_t stream) {
    (void)in_sizes; (void)n_in; (void)out_size; (void)d_ws; (void)ws_size;
    const float* Q  = (const float*)d_in[0];
    const float* Kp = (const float*)d_in[1];
    const float* V  = (const float*)d_in[2];
    const int*   Mk = (const int*)d_in[3];
    float* Out  = (float*)d_out;                       // [B,S,D]
    float* Attn = Out + (size_t)B_ * S_ * D_;          // [B,S,S]

    dim3 grid((NQT + WAVES - 1) / WAVES, B_);
    dim3 block(WAVES * 32);
    attn_tanh_wmma_kernel<<<grid, block, 0, stream>>>(Q, Kp, V, Mk, Out, Attn);
}